// MultiFrameLinearAttentionBlock_36524401885550
// MI455X (gfx1250) — compile-verified
//
#include <hip/hip_runtime.h>
#include <math.h>

typedef float v2f __attribute__((ext_vector_type(2)));
typedef float v4f __attribute__((ext_vector_type(4)));
typedef float v8f __attribute__((ext_vector_type(8)));

// Problem shape (fixed by reference setup_inputs)
constexpr int N_      = 8;
constexpr int C_      = 256;
constexpr int NF      = 16;
constexpr int WH      = 28 * 28;   // 784
constexpr int CS      = NF * WH;   // channel stride in l (elements) = 12544
constexpr int THREADS = 256;       // 8 waves (wave32)

__global__ __launch_bounds__(THREADS)
void mfla_fused(const float* __restrict__ l,   // [N,C,NF,W,H]
                const float* __restrict__ g,   // [N,C]
                const float* __restrict__ w,   // [C]
                float* __restrict__ out_c,     // [N,NF,WH]
                float* __restrict__ out_g)     // [N,C,NF]
{
    __shared__ float w_lds[C_];
    __shared__ float a_lds[WH];       // holds c, then a (in place)
    __shared__ float red[THREADS];

    const int tid  = threadIdx.x;
    const int lane = tid & 31;
    const int wave = tid >> 5;
    const int blk  = blockIdx.x;      // = n*NF + f
    const int n    = blk / NF;
    const int f    = blk - n * NF;

    const float* lbase = l + (size_t)n * C_ * CS + (size_t)f * WH;

    // ---- stage w into LDS; gw = dot(g[n,:], w) (softmax-invariant scalar) ----
    const float wv = w[tid];
    w_lds[tid] = wv;
    red[tid]   = wv * g[n * C_ + tid];
    __syncthreads();
    for (int s = THREADS / 2; s > 0; s >>= 1) {
        if (tid < s) red[tid] += red[tid + s];
        __syncthreads();
    }
    const float gw = red[0];
    __syncthreads();

    // ---- pass 1: c[wh] = dot(l[:,wh], w) + gw  (float4 over wh, coalesced) ----
    if (tid < WH / 4) {
        const float* lp = lbase + tid * 4;
        v4f acc = {0.f, 0.f, 0.f, 0.f};
        #pragma unroll 4
        for (int c = 0; c < C_; ++c) {
            const v4f lv = *(const v4f*)(lp + (size_t)c * CS);
            acc += lv * w_lds[c];
        }
        acc += gw;
        *(v4f*)(out_c + (size_t)blk * WH + tid * 4) = acc;
        *(v4f*)(&a_lds[tid * 4]) = acc;
    }
    __syncthreads();

    // ---- softmax over WH=784 (LDS tree reductions) ----
    float m = -INFINITY;
    for (int i = tid; i < WH; i += THREADS) m = fmaxf(m, a_lds[i]);
    red[tid] = m;
    __syncthreads();
    for (int s = THREADS / 2; s > 0; s >>= 1) {
        if (tid < s) red[tid] = fmaxf(red[tid], red[tid + s]);
        __syncthreads();
    }
    const float cmax = red[0];
    __syncthreads();

    float sum = 0.f;
    for (int i = tid; i < WH; i += THREADS) sum += __expf(a_lds[i] - cmax);
    red[tid] = sum;
    __syncthreads();
    for (int s = THREADS / 2; s > 0; s >>= 1) {
        if (tid < s) red[tid] += red[tid + s];
        __syncthreads();
    }
    const float inv_sum = 1.0f / red[0];
    __syncthreads();

    for (int i = tid; i < WH; i += THREADS)
        a_lds[i] = __expf(a_lds[i] - cmax) * inv_sum;
    __syncthreads();

    // ---- pass 2: g_out[ch] = sum_wh a[wh]*l[ch,wh] via V_WMMA_F32_16X16X4_F32 ----
    // A (16x4 f32): lane holds float2; m = lane&15; k-pair = (lane>>4)*2.
    // B (4x16 f32): rank-1 broadcast of a[] over N columns -> same k-pair per half-wave.
    //   B is identical for both channel groups of this wave, so fetch it ONCE per
    //   k-tile and feed two WMMAs (two 16-channel groups) from it.
    // D (16x16 f32): every column identical; lanes 0 / 16 own M=0..7 / M=8..15.
    const int mrow = lane & 15;
    const int kk   = (lane >> 4) << 1;       // 0 or 2
    const int ch0  = wave * 32;              // 8 waves x 32 channels = 256
    const float* ap0 = lbase + (size_t)(ch0 + mrow) * CS + kk;
    const float* ap1 = ap0 + (size_t)16 * CS;

    v8f acc0 = {};
    v8f acc1 = {};
    #pragma unroll 2
    for (int kt = 0; kt < WH; kt += 4) {
        v2f bv;
        bv.x = a_lds[kt + kk];
        bv.y = a_lds[kt + kk + 1];
        const v2f a0 = *(const v2f*)(ap0 + kt);
        const v2f a1 = *(const v2f*)(ap1 + kt);
        acc0 = __builtin_amdgcn_wmma_f32_16x16x4_f32(
            false, a0, false, bv, (short)0, acc0, false, false);
        acc1 = __builtin_amdgcn_wmma_f32_16x16x4_f32(
            false, a1, false, bv, (short)0, acc1, false, false);
    }

    if (mrow == 0) {
        const int half = (lane >> 4) * 8;    // VGPR j holds M=j (+8 for hi half)
        float* gp0 = out_g + ((size_t)n * C_ + ch0 + half) * NF + f;
        float* gp1 = gp0 + (size_t)16 * NF;
        #pragma unroll
        for (int j = 0; j < 8; ++j) gp0[(size_t)j * NF] = acc0[j];
        #pragma unroll
        for (int j = 0; j < 8; ++j) gp1[(size_t)j * NF] = acc1[j];
    }
}

extern "C" void kernel_launch(void* const* d_in, const int* in_sizes, int n_in,
                              void* d_out, int out_size, void* d_ws, size_t ws_size,
                              hipStream_t stream) {
    const float* l = (const float*)d_in[0];
    const float* g = (const float*)d_in[1];
    const float* w = (const float*)d_in[2];
    float* out   = (float*)d_out;
    float* out_c = out;                        // [N,NF,WH] = 100352 floats
    float* out_g = out + (size_t)N_ * NF * WH; // [N,C,NF]  = 32768 floats

    mfla_fused<<<N_ * NF, THREADS, 0, stream>>>(l, g, w, out_c, out_g);
}